// OnnxMultiHeadedAttention_21749714387025
// MI455X (gfx1250) — compile-verified
//
#include <hip/hip_runtime.h>
#include <hip/hip_bf16.h>

// ---------------------------------------------------------------------------
// MultiHeadedAttention, fp32, CDNA5 (gfx1250), wave32, V_WMMA_F32_16X16X4_F32
// B=2, S=4096, NF=512, H=8, D=64
// ---------------------------------------------------------------------------

typedef float v2f __attribute__((ext_vector_type(2)));
typedef float v8f __attribute__((ext_vector_type(8)));

#define S_LEN   4096
#define NFEAT   512
#define NHEAD   8
#define DK      64
#define BATCH   2
#define MTOT    (BATCH * S_LEN)      // 8192 rows

// ----------------------------- GEMM core -----------------------------------
// C[M x N] = A[M x 512] @ W[512 x N-block] + bias, tile 128(M) x 64(N) per
// 256-thread block; 8 waves in 4(M) x 2(N); each wave 32x32 (2x2 WMMA tiles).
// PERMUTE=true scatters output to [B, H, S, 64] (QKV); else row-major [M,512].
template <bool PERMUTE>
__device__ __forceinline__ void gemm_core(const float* __restrict__ A,
                                          const float* __restrict__ W,
                                          const float* __restrict__ bias,
                                          float* __restrict__ out)
{
    __shared__ float As[128][20];   // pad 20: 80B rows -> float4-aligned, no bank conflicts
    __shared__ float Ws[64][18];    // W^T tile, pad 18: 72B rows -> float2-aligned

    const int tid  = threadIdx.x;
    const int lane = tid & 31;
    const int wave = tid >> 5;
    const int l16  = lane & 15;
    const int hh   = lane >> 4;          // half-wave select (K/M-offset per WMMA layout)
    const int m0   = blockIdx.y * 128;
    const int n0   = blockIdx.x * 64;
    const int mw   = (wave >> 1) * 32;
    const int nw   = (wave & 1) * 32;

    v8f acc[2][2] = {};

    for (int k0 = 0; k0 < NFEAT; k0 += 16) {
        __syncthreads();
        // Stage A 128x16 (512 float4 loads over 256 threads)
        #pragma unroll
        for (int i = 0; i < 2; ++i) {
            int idx = tid + 256 * i;           // 0..511
            int row = idx >> 2;                // 0..127
            int c4  = (idx & 3) * 4;           // 0,4,8,12
            float4 va = *(const float4*)(A + (size_t)(m0 + row) * NFEAT + k0 + c4);
            *(float4*)(&As[row][c4]) = va;
        }
        // Stage W^T: rows k0..k0+15, cols n0..n0+63 -> Ws[n][k]
        {
            int row = tid >> 4;                // 0..15 (k)
            int c4  = (tid & 15) * 4;          // 0..60 (n)
            float4 vw = *(const float4*)(W + (size_t)(k0 + row) * NFEAT + n0 + c4);
            Ws[c4 + 0][row] = vw.x;  Ws[c4 + 1][row] = vw.y;
            Ws[c4 + 2][row] = vw.z;  Ws[c4 + 3][row] = vw.w;
        }
        __syncthreads();

        #pragma unroll
        for (int kk = 0; kk < 16; kk += 4) {
            v2f a[2], b[2];
            #pragma unroll
            for (int mt = 0; mt < 2; ++mt)   // A-layout: lane<16 K=0..1, lane>=16 K=2..3
                a[mt] = *(const v2f*)(&As[mw + 16 * mt + l16][kk + 2 * hh]);
            #pragma unroll
            for (int nt = 0; nt < 2; ++nt)   // B-layout: lane holds column n, K pair by half
                b[nt] = *(const v2f*)(&Ws[nw + 16 * nt + l16][kk + 2 * hh]);
            #pragma unroll
            for (int mt = 0; mt < 2; ++mt)
                #pragma unroll
                for (int nt = 0; nt < 2; ++nt)
                    acc[mt][nt] = __builtin_amdgcn_wmma_f32_16x16x4_f32(
                        false, a[mt], false, b[nt], (short)0, acc[mt][nt], false, false);
        }
    }

    // Epilogue: bias + store.  C-layout: vgpr r -> row r + 8*hh, col = lane%16.
    #pragma unroll
    for (int mt = 0; mt < 2; ++mt) {
        #pragma unroll
        for (int nt = 0; nt < 2; ++nt) {
            int n = n0 + nw + 16 * nt + l16;
            float bv = bias[n];
            #pragma unroll
            for (int r = 0; r < 8; ++r) {
                int m = m0 + mw + 16 * mt + r + 8 * hh;
                float val = acc[mt][nt][r] + bv;
                if (PERMUTE) {
                    int b_ = m >> 12;          // m / 4096
                    int s_ = m & (S_LEN - 1);
                    int h_ = n >> 6;           // n / 64
                    int d_ = n & (DK - 1);
                    out[(((size_t)(b_ * NHEAD + h_) * S_LEN) + s_) * DK + d_] = val;
                } else {
                    out[(size_t)m * NFEAT + n] = val;
                }
            }
        }
    }
}

__global__ __launch_bounds__(256) void qkv_kernel(
    const float* __restrict__ x,
    const float* __restrict__ wq, const float* __restrict__ bq,
    const float* __restrict__ wk, const float* __restrict__ bk,
    const float* __restrict__ wv, const float* __restrict__ bv,
    float* __restrict__ q, float* __restrict__ k, float* __restrict__ v)
{
    const float* w; const float* bias; float* o;
    if (blockIdx.z == 0)      { w = wq; bias = bq; o = q; }
    else if (blockIdx.z == 1) { w = wk; bias = bk; o = k; }
    else                      { w = wv; bias = bv; o = v; }
    gemm_core<true>(x, w, bias, o);
}

__global__ __launch_bounds__(256) void proj_kernel(
    const float* __restrict__ ctx,
    const float* __restrict__ wo, const float* __restrict__ bo,
    float* __restrict__ out)
{
    gemm_core<false>(ctx, wo, bo, out);
}

// --------------------------- Flash attention --------------------------------
// Block: 128 query rows of one (b,h); 8 waves x 16 rows. Streams keys in
// blocks of 32 with K/V^T staged in LDS, online softmax, WMMA f32 16x16x4.
__global__ __launch_bounds__(256) void flash_attn_kernel(
    const float* __restrict__ Q,     // [B*H][S][64]
    const float* __restrict__ K,
    const float* __restrict__ V,
    const float* __restrict__ mask,  // [B][S] (broadcast additive)
    float* __restrict__ ctx)         // [B][S][H][64]
{
    __shared__ float Ks[32][68];         // pad 68 -> 272B rows: float4-aligned, conflict-free
    __shared__ float Vt[64][34];         // V transposed [d][t], pad 34 -> 136B rows
    __shared__ float Ps[8][16][34];      // per-wave P tile (C-layout -> A-layout round trip)

    const int tid  = threadIdx.x;
    const int lane = tid & 31;
    const int wave = tid >> 5;
    const int l16  = lane & 15;
    const int hh   = lane >> 4;
    const int bh   = blockIdx.y;                    // 0..15
    const int b    = bh >> 3;
    const int h    = bh & (NHEAD - 1);
    const int m0   = blockIdx.x * 128 + wave * 16;  // this wave's query-row base

    const float* Qh = Q + (size_t)bh * S_LEN * DK;
    const float* Kh = K + (size_t)bh * S_LEN * DK;
    const float* Vh = V + (size_t)bh * S_LEN * DK;

    // Resident Q fragments (A-layout): lane<16 -> K=4k..4k+1, lane>=16 -> 4k+2..4k+3
    v2f qf[16];
    #pragma unroll
    for (int ks = 0; ks < 16; ++ks)
        qf[ks] = *(const v2f*)(Qh + (size_t)(m0 + l16) * DK + 4 * ks + 2 * hh);

    v8f acc[4] = {};                 // 16 x 64 ctx accumulator (4 N-tiles of d)
    float mrow[8], lrow[8];
    #pragma unroll
    for (int r = 0; r < 8; ++r) { mrow[r] = -1e30f; lrow[r] = 0.0f; }

    for (int t0 = 0; t0 < S_LEN; t0 += 32) {
        __syncthreads();
        // Stage K block (row-major) and V block (transposed) into LDS.
        #pragma unroll
        for (int i = 0; i < 2; ++i) {
            int idx = tid + 256 * i;          // 0..511
            int row = idx >> 4;               // 0..31 (key)
            int c4  = (idx & 15) * 4;         // 0..60 (d)
            float4 kv = *(const float4*)(Kh + (size_t)(t0 + row) * DK + c4);
            *(float4*)(&Ks[row][c4]) = kv;
            float4 vv = *(const float4*)(Vh + (size_t)(t0 + row) * DK + c4);
            Vt[c4 + 0][row] = vv.x;  Vt[c4 + 1][row] = vv.y;
            Vt[c4 + 2][row] = vv.z;  Vt[c4 + 3][row] = vv.w;
        }
        if (t0 + 32 < S_LEN) {               // uniform branch; hint next block into cache
            __builtin_prefetch(Kh + (size_t)(t0 + 32) * DK + lane * 16, 0, 0);
            __builtin_prefetch(Vh + (size_t)(t0 + 32) * DK + lane * 16, 0, 0);
        }
        __syncthreads();

        // S = Q @ K^T : two 16x16 tiles (keys t0..+15, t0+16..+31)
        v8f s[2] = {};
        #pragma unroll
        for (int j = 0; j < 2; ++j) {
            #pragma unroll
            for (int ks = 0; ks < 16; ++ks) {
                v2f bf = *(const v2f*)(&Ks[16 * j + l16][4 * ks + 2 * hh]);
                s[j] = __builtin_amdgcn_wmma_f32_16x16x4_f32(
                    false, qf[ks], false, bf, (short)0, s[j], false, false);
            }
        }

        // scale (1/sqrt(64)) + additive mask (per key column)
        float mv0 = mask[b * S_LEN + t0 + l16];
        float mv1 = mask[b * S_LEN + t0 + 16 + l16];
        #pragma unroll
        for (int r = 0; r < 8; ++r) {
            s[0][r] = s[0][r] * 0.125f + mv0;
            s[1][r] = s[1][r] * 0.125f + mv1;
        }

        // Online softmax: row r lives on 16 lanes of this half-wave.
        #pragma unroll
        for (int r = 0; r < 8; ++r) {
            float mx = fmaxf(s[0][r], s[1][r]);
            #pragma unroll
            for (int off = 8; off >= 1; off >>= 1)
                mx = fmaxf(mx, __shfl_xor(mx, off, 32));
            float mnew  = fmaxf(mrow[r], mx);
            float alpha = __expf(mrow[r] - mnew);
            float p0 = __expf(s[0][r] - mnew);
            float p1 = __expf(s[1][r] - mnew);
            s[0][r] = p0; s[1][r] = p1;
            float rs = p0 + p1;
            #pragma unroll
            for (int off = 8; off >= 1; off >>= 1)
                rs += __shfl_xor(rs, off, 32);
            lrow[r] = lrow[r] * alpha + rs;
            mrow[r] = mnew;
            #pragma unroll
            for (int j = 0; j < 4; ++j)
                acc[j][r] *= alpha;
        }

        // P: C-layout -> LDS -> A-layout (per-wave region, no barrier needed)
        #pragma unroll
        for (int r = 0; r < 8; ++r) {
            Ps[wave][r + 8 * hh][l16]      = s[0][r];
            Ps[wave][r + 8 * hh][16 + l16] = s[1][r];
        }

        // ctx += P(16x32) @ V(32x64)
        #pragma unroll
        for (int ks = 0; ks < 8; ++ks) {
            v2f af = *(const v2f*)(&Ps[wave][l16][4 * ks + 2 * hh]);
            #pragma unroll
            for (int j = 0; j < 4; ++j) {
                v2f bf = *(const v2f*)(&Vt[16 * j + l16][4 * ks + 2 * hh]);
                acc[j] = __builtin_amdgcn_wmma_f32_16x16x4_f32(
                    false, af, false, bf, (short)0, acc[j], false, false);
            }
        }
    }

    // Normalize and scatter ctx to [B, S, H, 64] (ready for out-projection GEMM)
    #pragma unroll
    for (int r = 0; r < 8; ++r) {
        int   srow = m0 + r + 8 * hh;
        float inv  = 1.0f / lrow[r];
        #pragma unroll
        for (int j = 0; j < 4; ++j) {
            int d = 16 * j + l16;
            ctx[((size_t)(b * S_LEN + srow) * NHEAD + h) * DK + d] = acc[j][r] * inv;
        }
    }
}

// ------------------------------- launcher -----------------------------------
extern "C" void kernel_launch(void* const* d_in, const int* in_sizes, int n_in,
                              void* d_out, int out_size, void* d_ws, size_t ws_size,
                              hipStream_t stream) {
    (void)in_sizes; (void)n_in; (void)out_size; (void)ws_size;
    const float* x    = (const float*)d_in[0];
    const float* mask = (const float*)d_in[1];
    const float* wq   = (const float*)d_in[2];
    const float* bq   = (const float*)d_in[3];
    const float* wk   = (const float*)d_in[4];
    const float* bk   = (const float*)d_in[5];
    const float* wv   = (const float*)d_in[6];
    const float* bv   = (const float*)d_in[7];
    const float* wo   = (const float*)d_in[8];
    const float* bo   = (const float*)d_in[9];
    float* out = (float*)d_out;

    const size_t SZ = (size_t)MTOT * NFEAT;   // 8192*512 floats per tensor
    float* q   = (float*)d_ws;
    float* k   = q + SZ;
    float* v   = k + SZ;
    float* ctx = v + SZ;

    dim3 blk(256);
    dim3 gQKV(NFEAT / 64, MTOT / 128, 3);     // 8 x 64 x 3
    qkv_kernel<<<gQKV, blk, 0, stream>>>(x, wq, bq, wk, bk, wv, bv, q, k, v);

    dim3 gAtt(S_LEN / 128, BATCH * NHEAD, 1); // 32 x 16
    flash_attn_kernel<<<gAtt, blk, 0, stream>>>(q, k, v, mask, ctx);

    dim3 gPrj(NFEAT / 64, MTOT / 128, 1);     // 8 x 64
    proj_kernel<<<gPrj, blk, 0, stream>>>(ctx, wo, bo, out);
}